// MultiHeadPool_25202868093582
// MI455X (gfx1250) — compile-verified
//
#include <hip/hip_runtime.h>

typedef __attribute__((ext_vector_type(2))) float v2f;
typedef __attribute__((ext_vector_type(8))) float v8f;

#define BB    8
#define NN    16384
#define MD    256
#define FF    32
#define HH    4
#define GG    32
#define G3C   32768
#define MCH   140      /* H*(F+3) real channels */
#define MPAD  144      /* padded to 9 WMMA tiles of 16 */
#define WAVES 9
#define TPB   (WAVES * 32)

// d_ws float layout:
//   [0:144)    channel sums
//   [144:288)  channel sum-of-squares
//   [288:432)  BN scale
//   [432:576)  BN shift
//   [576:576+144*256) zero-padded W (144 x 256)
#define WS_WPAD 576

// ---------------- Prep: zero-padded copy of W (kills divergent A-loads) ----------------
__global__ void prep_pad_w(const float* __restrict__ W, float* __restrict__ wpad)
{
  const int idx = blockIdx.x * 256 + threadIdx.x;   // 144*256 elements
  const int row = idx >> 8;                          // /256
  wpad[idx] = (row < MCH) ? W[idx] : 0.0f;
}

// ---------------- Pass A: WMMA GEMM + per-channel sum / sumsq ----------------
__global__ __launch_bounds__(TPB) void passA_gemm_stats(
    const float* __restrict__ X, const float* __restrict__ Wp,
    float* __restrict__ stats)
{
  const int wave = threadIdx.x >> 5;
  const int lane = threadIdx.x & 31;
  const int half = lane >> 4;       // which 16-lane half
  const int l15  = lane & 15;
  const int b    = blockIdx.x >> 6; // 64 chunks of 256 points per batch
  const int nb   = blockIdx.x & 63;
  const int nbase = nb * 256;
  const float* Xb = X + (size_t)b * MD * NN;
  const float* Arow = Wp + (size_t)(wave * 16 + l15) * MD + 2 * half;

  float s[8], q[8];
#pragma unroll
  for (int v = 0; v < 8; ++v) { s[v] = 0.f; q[v] = 0.f; }

  for (int sub = 0; sub < 16; ++sub) {
    const int n0 = nbase + sub * 16;
    const float* Bcol = Xb + (size_t)(2 * half) * NN + n0 + l15;
    v8f c = {};
    for (int k0 = 0; k0 < MD; k0 += 4) {
      v2f a, bm;
      a.x = Arow[k0];
      a.y = Arow[k0 + 1];
      bm.x = Bcol[(size_t)k0 * NN];
      bm.y = Bcol[(size_t)(k0 + 1) * NN];
      c = __builtin_amdgcn_wmma_f32_16x16x4_f32(false, a, false, bm,
                                                (short)0, c, false, false);
    }
#pragma unroll
    for (int v = 0; v < 8; ++v) { float y = c[v]; s[v] += y; q[v] += y * y; }
  }

  // reduce over the 16 lanes that share a channel, then one atomic per channel
#pragma unroll
  for (int v = 0; v < 8; ++v) {
    float sv = s[v], qv = q[v];
    for (int off = 8; off > 0; off >>= 1) {
      sv += __shfl_xor(sv, off, 16);
      qv += __shfl_xor(qv, off, 16);
    }
    if (l15 == 0) {
      const int m = wave * 16 + v + 8 * half;
      if (m < MCH) {
        atomicAdd(&stats[m], sv);
        atomicAdd(&stats[MPAD + m], qv);
      }
    }
  }
}

// ---------------- Finalize: BN scale/shift per channel ----------------
__global__ void finalize_bn(float* __restrict__ ws,
                            const float* __restrict__ kg, const float* __restrict__ kb,
                            const float* __restrict__ vg, const float* __restrict__ vb)
{
  const int c = threadIdx.x;
  if (c >= MPAD) return;
  float scale = 0.f, shift = 0.f;
  if (c < MCH) {
    const float inv_cnt = 1.0f / (float)(BB * NN);
    const float mean = ws[c] * inv_cnt;
    const float var  = ws[MPAD + c] * inv_cnt - mean * mean;
    const float rsig = rsqrtf(var + 1e-5f);
    float g, be;
    if (c < HH * 3) { g = kg[c];       be = kb[c]; }
    else            { g = vg[c - 12];  be = vb[c - 12]; }
    scale = g * rsig;
    shift = be - mean * scale;
  }
  ws[2 * MPAD + c] = scale;
  ws[3 * MPAD + c] = shift;
}

// ---------------- Pass B: WMMA GEMM + BN + transform + trilinear splat ----------------
__global__ __launch_bounds__(TPB) void passB_gemm_splat(
    const float* __restrict__ X, const float* __restrict__ P,
    const float* __restrict__ Wp, const float* __restrict__ ws,
    const float* __restrict__ Rm, const float* __restrict__ tv,
    float* __restrict__ out)
{
  __shared__ float kv[MPAD][16];  // normalized channels x 16 points
  __shared__ float cw[64][8];     // (head*16+point) x corner weight
  __shared__ int   ci[64][8];     // (head*16+point) x corner flat index

  const int wave = threadIdx.x >> 5;
  const int lane = threadIdx.x & 31;
  const int half = lane >> 4;
  const int l15  = lane & 15;
  const int b    = blockIdx.x >> 10; // 1024 groups of 16 points per batch
  const int nb   = blockIdx.x & 1023;
  const int n0   = nb * 16;
  const float* Xb = X + (size_t)b * MD * NN;
  const float* Arow = Wp + (size_t)(wave * 16 + l15) * MD + 2 * half;
  const float* Bcol = Xb + (size_t)(2 * half) * NN + n0 + l15;

  v8f c = {};
  for (int k0 = 0; k0 < MD; k0 += 4) {
    v2f a, bm;
    a.x = Arow[k0];
    a.y = Arow[k0 + 1];
    bm.x = Bcol[(size_t)k0 * NN];
    bm.y = Bcol[(size_t)(k0 + 1) * NN];
    c = __builtin_amdgcn_wmma_f32_16x16x4_f32(false, a, false, bm,
                                              (short)0, c, false, false);
  }

  const float* scale = ws + 2 * MPAD;
  const float* shift = ws + 3 * MPAD;
#pragma unroll
  for (int v = 0; v < 8; ++v) {
    const int m = wave * 16 + v + 8 * half;      // < 144 always
    kv[m][l15] = c[v] * scale[m] + shift[m];     // padded rows get 0
  }
  __syncthreads();

  // per (point, head): affine transform + tanh + corner weights/indices
  if (threadIdx.x < 64) {
    const int ph = threadIdx.x;
    const int h = ph >> 4, p = ph & 15;
    const int n = n0 + p;
    const float px = P[((size_t)b * 3 + 0) * NN + n] + kv[h * 3 + 0][p];
    const float py = P[((size_t)b * 3 + 1) * NN + n] + kv[h * 3 + 1][p];
    const float pz = P[((size_t)b * 3 + 2) * NN + n] + kv[h * 3 + 2][p];
    float loc[3]; int id[3];
#pragma unroll
    for (int i = 0; i < 3; ++i) {
      const float tk = Rm[h * 9 + i * 3 + 0] * px + Rm[h * 9 + i * 3 + 1] * py +
                       Rm[h * 9 + i * 3 + 2] * pz + tv[h * 3 + i];
      const float lat = tanhf(tk);
      const float pos = (lat + 1.0f) * (0.5f * (GG - 1));
      float fi = floorf(pos);
      fi = fminf(fmaxf(fi, 0.f), (float)(GG - 2));
      id[i]  = (int)fi;
      loc[i] = pos - fi;
    }
#pragma unroll
    for (int cor = 0; cor < 8; ++cor) {
      const int cx = (cor >> 2) & 1, cy = (cor >> 1) & 1, cz = cor & 1;
      const float w = (cx ? loc[0] : 1.f - loc[0]) *
                      (cy ? loc[1] : 1.f - loc[1]) *
                      (cz ? loc[2] : 1.f - loc[2]);
      cw[ph][cor] = w;
      ci[ph][cor] = ((id[0] + cx) * GG + (id[1] + cy)) * GG + (id[2] + cz);
    }
  }
  __syncthreads();

  // cooperative splat: 2048 (point, head, feature) units, 8 corner atomics each
  for (int u = threadIdx.x; u < 16 * HH * FF; u += TPB) {
    const int f  = u & 31;
    const int ph = u >> 5;
    const int h = ph >> 4, p = ph & 15;
    const float val = kv[12 + h * FF + f][p];
    float* obase = out + (((size_t)b * HH + h) * FF + f) * (size_t)G3C;
#pragma unroll
    for (int cor = 0; cor < 8; ++cor)
      atomicAdd(obase + ci[ph][cor], val * cw[ph][cor]);
  }
}

extern "C" void kernel_launch(void* const* d_in, const int* in_sizes, int n_in,
                              void* d_out, int out_size, void* d_ws, size_t ws_size,
                              hipStream_t stream) {
  const float* X  = (const float*)d_in[0];  // (B, 256, N)
  const float* P  = (const float*)d_in[1];  // (B, 3, N)
  const float* W  = (const float*)d_in[2];  // (140, 256)
  const float* kg = (const float*)d_in[3];
  const float* kb = (const float*)d_in[4];
  const float* vg = (const float*)d_in[5];
  const float* vb = (const float*)d_in[6];
  const float* Rm = (const float*)d_in[7];  // (4,3,3)
  const float* tv = (const float*)d_in[8];  // (4,3)
  float* out = (float*)d_out;
  float* ws  = (float*)d_ws;

  hipMemsetAsync(ws, 0, 2 * MPAD * sizeof(float), stream);          // zero stats
  hipMemsetAsync(out, 0, (size_t)out_size * sizeof(float), stream); // zero grid

  prep_pad_w<<<MPAD, 256, 0, stream>>>(W, ws + WS_WPAD);
  passA_gemm_stats<<<BB * (NN / 256), TPB, 0, stream>>>(X, ws + WS_WPAD, ws);
  finalize_bn<<<1, 160, 0, stream>>>(ws, kg, kb, vg, vb);
  passB_gemm_splat<<<BB * (NN / 16), TPB, 0, stream>>>(X, P, ws + WS_WPAD, ws,
                                                       Rm, tv, out);
}